// TemporalAttentionPooling_36086315221125
// MI455X (gfx1250) — compile-verified
//
#include <hip/hip_runtime.h>
#include <math.h>
#include <stdint.h>

// ---- problem constants ----
static constexpr int kB = 32;
static constexpr int kL = 4096;
static constexpr int kD = 1024;
static constexpr int kS = 8;                  // splits along L per batch
static constexpr int kChunk = kL / kS;        // 512 rows per workgroup
static constexpr int kTile = 16;              // rows per inner tile (WMMA M/K granularity)
static constexpr int kRow = kD + 16;          // padded LDS row stride (floats) -> bank-shift 32
static constexpr int kTiles = kChunk / kTile; // 32
static constexpr int kTileF = kTile * kRow;   // floats per LDS tile buffer
static constexpr int kTileBytes = kTileF * 4; // 66560 B

typedef float v2f __attribute__((ext_vector_type(2)));
typedef float v4f __attribute__((ext_vector_type(4)));
typedef float v8f __attribute__((ext_vector_type(8)));

// gfx1250 async copy: 16B/lane global -> LDS, tracked by ASYNCcnt.
// INST_OFFSET is added to BOTH the LDS and global address (ISA 15.18.3),
// so one base pair serves 4 issues covering 64 contiguous bytes.
template <int OFF>
__device__ __forceinline__ void async_ld_b128(uint32_t lds_off, uint64_t gaddr) {
    asm volatile("global_load_async_to_lds_b128 %0, %1, off offset:%2"
                 :: "v"(lds_off), "v"(gaddr), "i"(OFF) : "memory");
}
template <int N>
__device__ __forceinline__ void wait_asynccnt() {
#if defined(__has_builtin) && __has_builtin(__builtin_amdgcn_s_wait_asynccnt)
    __builtin_amdgcn_s_wait_asynccnt(N);
#else
    asm volatile("s_wait_asynccnt %0" :: "i"(N) : "memory");
#endif
}

// Pass 1: single-pass online-softmax attention pooling over an L-chunk.
// grid = (kB * kS), block = 1024 (32 wave32s), double-buffered async LDS staging.
__global__ __launch_bounds__(1024)
void tap_pass1(const float* __restrict__ seq,
               const unsigned char* __restrict__ mask,   // JAX bool: 1 byte/elem
               const float* __restrict__ query,
               float* __restrict__ ws_state,             // [kB*kS*2]  (m, denom)
               float* __restrict__ ws_acc)               // [kB*kS*kD]
{
    extern __shared__ float smem[];
    float* tiles = smem;                        // 2 * kTileF floats (double buffer)
    float* qs    = smem + 2 * kTileF;           // kD floats
    float* wts   = qs + kD;                     // kTile weights
    float* spart = wts + kTile;                 // 32 half-row dot partials
    float* st    = spart + 32;                  // [0]=f, [1]=m, [2]=denom (pad to 8)
    unsigned char* mk = (unsigned char*)(st + 8);  // kChunk mask bytes

    const int tid  = threadIdx.x;
    const int lane = tid & 31;
    const int wv   = tid >> 5;                  // 0..31
    const int b    = blockIdx.x / kS;
    const int sp   = blockIdx.x % kS;
    const int l0   = sp * kChunk;

    const int d0   = wv * 32;                   // this wave's 32 output columns
    const int coln = lane & 15;
    const int hi   = lane >> 4;                 // selects K-pair per WMMA layout
    const float scale = 0.03125f;               // 1/sqrt(1024)
    const float* seqb = seq + (size_t)b * kL * kD;

    // stage query + this chunk's mask bytes into LDS (once)
    for (int i = tid; i < kD; i += 1024) qs[i] = query[i];
    if (tid < kChunk) mk[tid] = mask[(size_t)b * kL + l0 + tid];
    if (tid == 0) { st[1] = -INFINITY; st[2] = 0.0f; }

    // per-thread staging coordinates: 64B of one row
    const int r  = tid >> 6;                    // 0..15 row in tile
    const int cb = (tid & 63) * 16;             // 0..1008 col base
    const uint32_t ldsLoc = (uint32_t)((r * kRow + cb) * 4);  // dynamic LDS base = 0
    uint64_t g = (uint64_t)(uintptr_t)(seqb + (size_t)(l0 + r) * kD + cb);
    const uint64_t gStep = (uint64_t)kTile * kD * 4;

    // prologue: async-load tile 0 into buffer 0
    async_ld_b128<0>(ldsLoc, g);
    async_ld_b128<16>(ldsLoc, g);
    async_ld_b128<32>(ldsLoc, g);
    async_ld_b128<48>(ldsLoc, g);
    g += gStep;

    v8f c0 = {}; v8f c1 = {};                   // WMMA accumulators (rows replicated)
    __syncthreads();                            // query/mask staged

    for (int t = 0; t < kTiles; ++t) {
        float* tile = tiles + (t & 1) * kTileF;

        // ---- issue async loads for tile t+1 into the other buffer ----
        if (t + 1 < kTiles) {
            const uint32_t nx = (uint32_t)(((t + 1) & 1) * kTileBytes) + ldsLoc;
            async_ld_b128<0>(nx, g);
            async_ld_b128<16>(nx, g);
            async_ld_b128<32>(nx, g);
            async_ld_b128<48>(nx, g);
            g += gStep;
            wait_asynccnt<4>();                 // tile t done; t+1 stays in flight
            if (t + 2 < kTiles)                 // nudge t+2 toward GL2
                __builtin_prefetch((const void*)(uintptr_t)g, 0, 2);
        } else {
            wait_asynccnt<0>();
        }
        __syncthreads();                        // all waves' tile-t portions visible

        // ---- scores: wave (wv&15) row, half (wv>>4); 512-elem dot per wave ----
        {
            const int rr = wv & 15;
            const int h  = wv >> 4;
            const float* rp = tile + rr * kRow + h * 512;
            const float* qp = qs + h * 512;
            float acc = 0.0f;
            #pragma unroll
            for (int j = 0; j < 4; ++j) {
                const int o = j * 128 + lane * 4;
                v4f x = *(const v4f*)(rp + o);
                v4f y = *(const v4f*)(qp + o);
                acc += x.x * y.x + x.y * y.y + x.z * y.z + x.w * y.w;
            }
            #pragma unroll
            for (int m = 16; m >= 1; m >>= 1)
                acc += __shfl_xor(acc, m, 32);
            if (lane == 0) spart[wv] = acc;
        }
        __syncthreads();

        // ---- online softmax state update: wave 0, lane-parallel ----
        if (wv == 0) {
            const float mo = st[1];
            const float dn = st[2];
            float v = -INFINITY;
            if (lane < kTile) {
                v = (spart[lane] + spart[lane + 16]) * scale;
                if (mk[t * kTile + lane] == 0) v = -INFINITY;
            }
            float cm = v;
            #pragma unroll
            for (int m = 16; m >= 1; m >>= 1)
                cm = fmaxf(cm, __shfl_xor(cm, m, 32));
            const float mn = fmaxf(mo, cm);
            float f, w;
            if (mn == -INFINITY) {              // everything masked so far
                f = 1.0f;
                w = 0.0f;
            } else {
                f = expf(mo - mn);              // exp(-inf)=0 on first live tile
                w = (lane < kTile) ? expf(v - mn) : 0.0f;
            }
            if (lane < kTile) wts[lane] = w;
            float wsum = w;
            #pragma unroll
            for (int m = 16; m >= 1; m >>= 1)
                wsum += __shfl_xor(wsum, m, 32);
            if (lane == 0) {
                st[0] = f;
                st[1] = mn;
                st[2] = dn * f + wsum;
            }
        }
        __syncthreads();

        // ---- rescale accumulators, then WMMA over the 16-row tile ----
        {
            const float f = st[0];
            c0 = c0 * f;
            c1 = c1 * f;
            #pragma unroll
            for (int k0 = 0; k0 < kTile; k0 += 4) {
                // A (16x4 f32): A[m,k] = wts[k0+k] broadcast over m.
                // lanes 0-15: K={0,1}; lanes 16-31: K={2,3}
                v2f a;
                a.x = wts[k0 + 2 * hi];
                a.y = wts[k0 + 2 * hi + 1];
                // B (4x16 f32): B[k,n] = tile[k0+k][d0+n]
                const float* bp = tile + (size_t)(k0 + 2 * hi) * kRow;
                v2f b0, b1;
                b0.x = bp[d0 + coln];
                b0.y = bp[kRow + d0 + coln];
                b1.x = bp[d0 + 16 + coln];
                b1.y = bp[kRow + d0 + 16 + coln];
                c0 = __builtin_amdgcn_wmma_f32_16x16x4_f32(
                        false, a, false, b0, (short)0, c0, false, false);
                c1 = __builtin_amdgcn_wmma_f32_16x16x4_f32(
                        false, a, false, b1, (short)0, c1, false, false);
            }
        }
        __syncthreads();    // frees tile buf for async overwrite at t+1's issue
    }

    // ---- write per-chunk partials ----
    if (tid == 0) {
        ws_state[(b * kS + sp) * 2 + 0] = st[1];
        ws_state[(b * kS + sp) * 2 + 1] = st[2];
    }
    if (lane < 16) {    // row 0 of D lives in c[0], lanes 0-15 (cols = lane)
        float* dst = ws_acc + (size_t)(b * kS + sp) * kD;
        dst[d0 + lane]      = c0[0];
        dst[d0 + 16 + lane] = c1[0];
    }
}

// Pass 2: merge kS online-softmax partials per batch. grid = kB, block = 256.
__global__ __launch_bounds__(256)
void tap_pass2(const float* __restrict__ ws_state,
               const float* __restrict__ ws_acc,
               float* __restrict__ out)
{
    const int b = blockIdx.x;
    float mg = -INFINITY;
    #pragma unroll
    for (int s = 0; s < kS; ++s)
        mg = fmaxf(mg, ws_state[(b * kS + s) * 2]);
    float sc[kS];
    float dn = 0.0f;
    #pragma unroll
    for (int s = 0; s < kS; ++s) {
        float e = (mg == -INFINITY) ? 0.0f : expf(ws_state[(b * kS + s) * 2] - mg);
        sc[s] = e;
        dn += e * ws_state[(b * kS + s) * 2 + 1];
    }
    const float inv = 1.0f / dn;
    for (int d = threadIdx.x; d < kD; d += 256) {
        float acc = 0.0f;
        #pragma unroll
        for (int s = 0; s < kS; ++s)
            acc += sc[s] * ws_acc[(size_t)(b * kS + s) * kD + d];
        out[(size_t)b * kD + d] = acc * inv;
    }
}

extern "C" void kernel_launch(void* const* d_in, const int* in_sizes, int n_in,
                              void* d_out, int out_size, void* d_ws, size_t ws_size,
                              hipStream_t stream) {
    const float* seq          = (const float*)d_in[0];
    const unsigned char* mask = (const unsigned char*)d_in[1];
    const float* query        = (const float*)d_in[2];
    float* out                = (float*)d_out;

    float* ws_state = (float*)d_ws;                    // kB*kS*2 floats
    float* ws_acc   = ws_state + kB * kS * 2;          // kB*kS*kD floats (~1 MB total)

    const size_t shmem = (size_t)(2 * kTileF + kD + kTile + 32 + 8) * sizeof(float)
                       + (size_t)kChunk;               // mask bytes

    tap_pass1<<<dim3(kB * kS), dim3(1024), shmem, stream>>>(seq, mask, query,
                                                            ws_state, ws_acc);
    tap_pass2<<<dim3(kB), dim3(256), 0, stream>>>(ws_state, ws_acc, out);
}